// GCN_64828236366125
// MI455X (gfx1250) — compile-verified
//
#include <hip/hip_runtime.h>

// ---------------------------------------------------------------------------
// GraphSAGE (mean) 3-layer inference for MI455X / gfx1250 (wave32, WMMA).
//
// Reformulation: mean_j(x_j) @ W == mean_j(x_j @ W)  -> project first with
// fp32 WMMA GEMMs, then scatter the *narrow* projected features over edges
// (64/32 floats per edge instead of 128/64) -> 2x less aggregation traffic.
// All hot tables (<= 12.8 MB) live in the 192 MB L2.
// ---------------------------------------------------------------------------

typedef __attribute__((ext_vector_type(2))) float v2f;
typedef __attribute__((ext_vector_type(8))) float v8f;

// Y[N,M] = X[N,K] @ W[K,M], row-major, fp32, one wave per 16x16 C-tile.
// Uses V_WMMA_F32_16X16X4_F32 (full-precision fp32 matrix core path).
// Fragment layouts per CDNA5 ISA 7.12.2:
//   A 16x4 : lane m = lane&15 holds A[m][kk], A[m][kk+1], kk = (lane>>4)*2
//   B 4x16 : lane n = lane&15 holds B[kk][n], B[kk+1][n]
//   C 16x16: vgpr r, lanes 0-15 -> row r, lanes 16-31 -> row r+8, col = lane&15
template <int K, int M>
__global__ void sage_gemm_wmma(const float* __restrict__ X,
                               const float* __restrict__ W,
                               float* __restrict__ Y, int tilesN) {
  const int lane = threadIdx.x & 31;
  const int wave = threadIdx.x >> 5;
  const int tN = blockIdx.x * (blockDim.x >> 5) + wave;  // 16-row tile index
  if (tN >= tilesN) return;                              // wave-uniform branch
  const int tM = blockIdx.y;                             // 16-col tile index

  const int m  = lane & 15;
  const int kk = (lane >> 4) << 1;                       // 0 or 2

  const float* __restrict__ xrow = X + (size_t)(tN * 16 + m) * K;
  const int wcol = tM * 16 + m;

  v8f c = {};
#pragma unroll
  for (int k0 = 0; k0 < K; k0 += 4) {
    v2f a, b;
    a.x = xrow[k0 + kk];
    a.y = xrow[k0 + kk + 1];
    b.x = W[(size_t)(k0 + kk)     * M + wcol];
    b.y = W[(size_t)(k0 + kk + 1) * M + wcol];
    // (neg_a, A, neg_b, B, c_mod, C, reuse_a, reuse_b)
    c = __builtin_amdgcn_wmma_f32_16x16x4_f32(false, a, false, b, (short)0, c,
                                              false, false);
  }

  const int col   = tM * 16 + (lane & 15);
  const int rbase = tN * 16 + ((lane >> 4) << 3);
#pragma unroll
  for (int r = 0; r < 8; ++r)
    Y[(size_t)(rbase + r) * M + col] = c[r];
}

// deg[i] = #incoming edges (as float)
__global__ void degree_kernel(const int* __restrict__ dst,
                              float* __restrict__ deg, int E) {
  int e = blockIdx.x * blockDim.x + threadIdx.x;
  if (e < E) atomicAdd(&deg[dst[e]], 1.0f);
}

// agg[dst[e], f] += y[src[e], f]; one thread per (edge, feature).
// F in {64,32}: each wave handles one edge's features -> coalesced gather,
// consecutive atomic addresses (spread across L2 channels).
template <int F>
__global__ void scatter_add(const float* __restrict__ y,
                            const int* __restrict__ src,
                            const int* __restrict__ dst,
                            float* __restrict__ agg, int E) {
  const int e = blockIdx.x * (blockDim.x / F) + threadIdx.x / F;
  const int f = threadIdx.x & (F - 1);
  if (e >= E) return;
  const int s = src[e];
  const int d = dst[e];
  atomicAdd(&agg[(size_t)d * F + f], y[(size_t)s * F + f]);
}

// h = relu(agg / max(deg,1) + r + bias)
template <int F>
__global__ void combine_relu(const float* __restrict__ agg,
                             const float* __restrict__ deg,
                             const float* __restrict__ r,
                             const float* __restrict__ bias,
                             float* __restrict__ h, int N) {
  const int idx = blockIdx.x * blockDim.x + threadIdx.x;
  if (idx >= N * F) return;
  const int n = idx / F;
  const int f = idx & (F - 1);
  const float d = fmaxf(deg[n], 1.0f);
  h[idx] = fmaxf(agg[idx] / d + r[idx] + bias[f], 0.0f);
}

// out[n, 0:2] = relu(h2[n, 0:32] @ w3 + b3)  (head is tiny -> plain VALU)
__global__ void head_kernel(const float* __restrict__ h2,
                            const float* __restrict__ w3,
                            const float* __restrict__ b3,
                            float* __restrict__ out, int N) {
  const int n = blockIdx.x * blockDim.x + threadIdx.x;
  if (n >= N) return;
  float acc0 = b3[0], acc1 = b3[1];
  const float* __restrict__ h = h2 + (size_t)n * 32;
#pragma unroll
  for (int k = 0; k < 32; ++k) {
    const float v = h[k];
    acc0 += v * w3[k * 2 + 0];
    acc1 += v * w3[k * 2 + 1];
  }
  out[n * 2 + 0] = fmaxf(acc0, 0.0f);
  out[n * 2 + 1] = fmaxf(acc1, 0.0f);
}

extern "C" void kernel_launch(void* const* d_in, const int* in_sizes, int n_in,
                              void* d_out, int out_size, void* d_ws,
                              size_t ws_size, hipStream_t stream) {
  (void)n_in; (void)out_size; (void)ws_size;

  const float* x    = (const float*)d_in[0];
  const int*   eidx = (const int*)d_in[1];   // [2, E] flattened: row0=src, row1=dst
  const float* w1_l = (const float*)d_in[2];
  const float* w1_r = (const float*)d_in[3];
  const float* b1   = (const float*)d_in[4];
  const float* w2_l = (const float*)d_in[5];
  const float* w2_r = (const float*)d_in[6];
  const float* b2   = (const float*)d_in[7];
  const float* w3   = (const float*)d_in[8];
  const float* b3   = (const float*)d_in[9];
  float* out = (float*)d_out;

  const int FIN = 128;
  const int N = in_sizes[0] / FIN;   // 50000 (= 3125 * 16)
  const int E = in_sizes[1] / 2;     // 1,600,000
  const int H1 = in_sizes[4];        // 64
  const int H2 = in_sizes[7];        // 32

  const int* src = eidx;
  const int* dst = eidx + E;

  // Workspace layout (floats); agg1/agg2/deg contiguous for one memset.
  float* ws = (float*)d_ws;
  size_t o = 0;
  float* y1   = ws + o; o += (size_t)N * H1;
  float* r1   = ws + o; o += (size_t)N * H1;
  float* h1   = ws + o; o += (size_t)N * H1;
  float* y2   = ws + o; o += (size_t)N * H2;
  float* r2   = ws + o; o += (size_t)N * H2;
  float* h2   = ws + o; o += (size_t)N * H2;
  float* agg1 = ws + o; o += (size_t)N * H1;
  float* agg2 = ws + o; o += (size_t)N * H2;
  float* deg  = ws + o; o += (size_t)N;

  // Zero the accumulators every call (atomics accumulate into them).
  hipMemsetAsync(agg1, 0, (size_t)N * (H1 + H2 + 1) * sizeof(float), stream);

  degree_kernel<<<(E + 255) / 256, 256, 0, stream>>>(dst, deg, E);

  const int tilesN = N / 16;  // 3125, exact
  dim3 blk(256);              // 8 wave32 waves; each wave = one 16x16 C tile

  // ---- Layer 1: project (WMMA fp32), aggregate 64-wide, combine ----
  dim3 g1((tilesN + 7) / 8, H1 / 16);
  sage_gemm_wmma<128, 64><<<g1, blk, 0, stream>>>(x, w1_l, y1, tilesN);
  sage_gemm_wmma<128, 64><<<g1, blk, 0, stream>>>(x, w1_r, r1, tilesN);

  {
    const long long tot = (long long)E * H1;
    scatter_add<64><<<(unsigned)((tot + 255) / 256), 256, 0, stream>>>(
        y1, src, dst, agg1, E);
    combine_relu<64><<<(N * H1 + 255) / 256, 256, 0, stream>>>(
        agg1, deg, r1, b1, h1, N);
  }

  // ---- Layer 2: project (WMMA fp32), aggregate 32-wide, combine ----
  dim3 g2((tilesN + 7) / 8, H2 / 16);
  sage_gemm_wmma<64, 32><<<g2, blk, 0, stream>>>(h1, w2_l, y2, tilesN);
  sage_gemm_wmma<64, 32><<<g2, blk, 0, stream>>>(h1, w2_r, r2, tilesN);

  {
    const long long tot = (long long)E * H2;
    scatter_add<32><<<(unsigned)((tot + 255) / 256), 256, 0, stream>>>(
        y2, src, dst, agg2, E);
    combine_relu<32><<<(N * H2 + 255) / 256, 256, 0, stream>>>(
        agg2, deg, r2, b2, h2, N);
  }

  // ---- Head: 32 -> 2 + relu ----
  head_kernel<<<(N + 255) / 256, 256, 0, stream>>>(h2, w3, b3, out, N);
}